// GVPMultiEdgeConv_14834817040845
// MI455X (gfx1250) — compile-verified
//
#include <hip/hip_runtime.h>
#include <hip/hip_bf16.h>

#define NA 100000
#define NB 50000
#define EE 250000

typedef __attribute__((ext_vector_type(2))) float v2f;
typedef __attribute__((ext_vector_type(8))) float v8f;

// ---------------------------------------------------------------------------
// Pack a row-major weight matrix W[Krows, Ncols] into WMMA-B fragment order:
// Bp[((ks*nTiles)+nt)*64 + lane*2 + j] = W[(ks*4 + j + 2*(lane>=16)) * Ncols
//                                          + nt*16 + (lane&15)]   (0 if K OOB)
// so the GEMM inner loop issues one coalesced 256B b64 load per (ks,nt).
// ---------------------------------------------------------------------------
__global__ void pack_b_kernel(const float* __restrict__ W, float* __restrict__ Bp,
                              int Krows, int Ncols, int nTiles, int kSteps) {
  int idx = blockIdx.x * blockDim.x + threadIdx.x;
  int total = kSteps * nTiles * 64;
  if (idx >= total) return;
  int ks  = idx / (nTiles * 64);
  int rem = idx % (nTiles * 64);
  int nt  = rem / 64;
  int q   = rem % 64;
  int lane = q >> 1, j = q & 1;
  int K = ks * 4 + j + 2 * (lane >> 4);
  int n = nt * 16 + (lane & 15);
  Bp[idx] = (K < Krows) ? W[K * Ncols + n] : 0.0f;
}

// ---------------------------------------------------------------------------
// Edge-message kernel: one wave per 16-edge tile. 2 waves / 64-thread block.
// Per tile: gather h/v/x, VALU GVP vector path (Vh, sh, Vu), then
//   feats = silu([h|sh] @ Wf + bf)   via v_wmma_f32_16x16x4_f32 (K=148 padded)
//   gate  = sigmoid(feats @ Wg + bg) via WMMA (K=128)
// scatter-add feats into s_sum and gate*Vu into v_sum with f32 atomics.
// ---------------------------------------------------------------------------
__global__ __launch_bounds__(64) void edge_kernel(
    const float* __restrict__ h_src, const float* __restrict__ v_src,
    const float* __restrict__ x_src, const float* __restrict__ x_dst,
    const int* __restrict__ src, const int* __restrict__ dst,
    const float* __restrict__ Wh, const float* __restrict__ Wu,
    const float* __restrict__ BpF, const float* __restrict__ bf,
    const float* __restrict__ BpG, const float* __restrict__ bg,
    float* __restrict__ s_sum, float* __restrict__ v_sum, int nTiles)
{
  extern __shared__ float smem[];
  float* Wh_s = smem;             // 17*17 = 289
  float* Wu_s = Wh_s + 289;       // 17*16 = 272
  float* base = Wu_s + 272;
  const int wv = threadIdx.x >> 5;
  float* wbase = base + wv * 4800;
  float* sA    = wbase;           // 16 x 148  (s = [h | sh | pad], later feats)
  float* vin_s = sA + 2368;       // 16 x 51
  float* vh_s  = vin_s + 816;     // 16 x 51
  float* vu_s  = vh_s + 816;      // 16 x 48
  int*   esrc  = (int*)(vu_s + 768); // 16
  int*   edst  = esrc + 16;          // 16

  for (int i = threadIdx.x; i < 289; i += 64) Wh_s[i] = Wh[i];
  for (int i = threadIdx.x; i < 272; i += 64) Wu_s[i] = Wu[i];

  const int lane = threadIdx.x & 31;
  const int lid = lane & 15, hf = lane >> 4;
  int tile = blockIdx.x * 2 + wv;
  const bool wvalid = tile < nTiles;
  if (!wvalid) tile = nTiles - 1;   // recompute-safe; atomics are masked
  const int e0 = tile * 16;

  if (hf == 0) { esrc[lid] = src[e0 + lid]; edst[lid] = dst[e0 + lid]; }
  __syncthreads();

  // gather scalar features (float4, coalesced across the wave per row)
  for (int m = 0; m < 16; ++m) {
    const float4 hv = *(const float4*)(h_src + (long)esrc[m] * 128 + lane * 4);
    float* p = sA + m * 148 + lane * 4;
    p[0] = hv.x; p[1] = hv.y; p[2] = hv.z; p[3] = hv.w;
  }
  if (hf == 0) { sA[lid*148+145] = 0.f; sA[lid*148+146] = 0.f; sA[lid*148+147] = 0.f; }

  // vin = [x_src - x_dst ; v_src]  (17 channels x 3)
  {
    const int sidx = esrc[lid], didx = edst[lid];
    const int t0 = hf ? 26 : 0, t1 = hf ? 51 : 26;
    for (int t = t0; t < t1; ++t) {
      int k = t / 3, c = t - k * 3;
      float val = (k == 0)
          ? (x_src[(long)sidx * 3 + c] - x_dst[(long)didx * 3 + c])
          : v_src[(long)sidx * 48 + (k - 1) * 3 + c];
      vin_s[lid * 51 + t] = val;
    }
  }
  __syncthreads();

  // Vh = Wh^T * vin ; sh = |Vh|_c   (half-waves split h)
  {
    const int h0 = hf ? 9 : 0, h1 = hf ? 17 : 9;
    for (int h = h0; h < h1; ++h) {
      float a0 = 0, a1 = 0, a2 = 0;
      for (int k = 0; k < 17; ++k) {
        float w = Wh_s[k * 17 + h];
        a0 += vin_s[lid*51 + k*3 + 0] * w;
        a1 += vin_s[lid*51 + k*3 + 1] * w;
        a2 += vin_s[lid*51 + k*3 + 2] * w;
      }
      vh_s[lid*51 + h*3 + 0] = a0; vh_s[lid*51 + h*3 + 1] = a1; vh_s[lid*51 + h*3 + 2] = a2;
      sA[lid*148 + 128 + h] = sqrtf(a0*a0 + a1*a1 + a2*a2);
    }
  }
  __syncthreads();

  // Vu = Wu^T * Vh
  for (int u = hf * 8; u < hf * 8 + 8; ++u) {
    float a0 = 0, a1 = 0, a2 = 0;
    for (int h = 0; h < 17; ++h) {
      float w = Wu_s[h * 16 + u];
      a0 += vh_s[lid*51 + h*3 + 0] * w;
      a1 += vh_s[lid*51 + h*3 + 1] * w;
      a2 += vh_s[lid*51 + h*3 + 2] * w;
    }
    vu_s[lid*48 + u*3 + 0] = a0; vu_s[lid*48 + u*3 + 1] = a1; vu_s[lid*48 + u*3 + 2] = a2;
  }
  __syncthreads();

  // feats GEMM: [16 x 148] @ [148 x 128], fp32 WMMA, 37 K-steps x 8 N-tiles
  v8f acc[8];
  for (int nt = 0; nt < 8; ++nt) {
    float b0 = bf[nt * 16 + lid];
    for (int i = 0; i < 8; ++i) acc[nt][i] = b0;
  }
  for (int ks = 0; ks < 37; ++ks) {
    v2f a;
    int kb = ks * 4 + 2 * hf;
    a[0] = sA[lid * 148 + kb]; a[1] = sA[lid * 148 + kb + 1];
    const float* bp = BpF + (long)ks * 512 + lane * 2;
    for (int nt = 0; nt < 8; ++nt) {
      v2f b; b[0] = bp[nt * 64]; b[1] = bp[nt * 64 + 1];
      acc[nt] = __builtin_amdgcn_wmma_f32_16x16x4_f32(false, a, false, b,
                                                      (short)0, acc[nt], false, false);
    }
  }
  __syncthreads();

  // silu, scatter to s_sum, keep feats in LDS for the gate GEMM
  for (int nt = 0; nt < 8; ++nt) {
    for (int r = 0; r < 8; ++r) {
      float x = acc[nt][r];
      float so = x / (1.0f + __expf(-x));
      int M = r + 8 * hf, N = nt * 16 + lid;
      sA[M * 148 + N] = so;
      if (wvalid) atomicAdd(s_sum + (long)edst[M] * 128 + N, so);
    }
  }
  __syncthreads();

  // gate GEMM: [16 x 128] @ [128 x 16]
  v8f g;
  { float b0 = bg[lid]; for (int i = 0; i < 8; ++i) g[i] = b0; }
  for (int ks = 0; ks < 32; ++ks) {
    v2f a;
    int kb = ks * 4 + 2 * hf;
    a[0] = sA[lid * 148 + kb]; a[1] = sA[lid * 148 + kb + 1];
    const float* bp = BpG + ks * 64 + lane * 2;
    v2f b; b[0] = bp[0]; b[1] = bp[1];
    g = __builtin_amdgcn_wmma_f32_16x16x4_f32(false, a, false, b, (short)0, g, false, false);
  }
  if (wvalid) {
    for (int r = 0; r < 8; ++r) {
      int M = r + 8 * hf;
      float gv = 1.0f / (1.0f + __expf(-g[r]));
      long vb = (long)edst[M] * 48 + lid * 3;
      atomicAdd(v_sum + vb + 0, gv * vu_s[M*48 + lid*3 + 0]);
      atomicAdd(v_sum + vb + 1, gv * vu_s[M*48 + lid*3 + 1]);
      atomicAdd(v_sum + vb + 2, gv * vu_s[M*48 + lid*3 + 2]);
    }
  }
}

// ---------------------------------------------------------------------------
// Node-update kernel: one wave per 16-node tile. Residual add, LN, vector
// norm, update GVP (WMMA GEMMs, K=144 exact and K=128), residual, final LN,
// write h / x / v to d_out.
// ---------------------------------------------------------------------------
__global__ __launch_bounds__(64) void node_kernel(
    const float* __restrict__ h_a, const float* __restrict__ x_a, const float* __restrict__ v_a,
    const float* __restrict__ s_sum, const float* __restrict__ v_sum,
    const float* __restrict__ Whu, const float* __restrict__ Wuu,
    const float* __restrict__ BpU, const float* __restrict__ bf_u,
    const float* __restrict__ BpGU, const float* __restrict__ bg_u,
    const float* __restrict__ g_msg, const float* __restrict__ b_msg,
    const float* __restrict__ g_upd, const float* __restrict__ b_upd,
    float* __restrict__ out, int nTiles)
{
  extern __shared__ float smem[];
  float* Whu_s = smem;            // 256
  float* Wuu_s = Whu_s + 256;     // 256
  float* base  = Wuu_s + 256;
  const int wv = threadIdx.x >> 5;
  float* wbase = base + wv * 6832;
  float* sA   = wbase;            // 16 x 148  (nf | sh_u)
  float* fo   = sA + 2368;        // 16 x 128  (feats_u)
  float* vc   = fo + 2048;        // 16 x 48   (v -> vln -> v2)
  float* vh_s = vc + 768;         // 16 x 48
  float* vu_s = vh_s + 768;       // 16 x 48
  float* red  = vu_s + 768;       // 64
  float* mu_s = red + 64;         // 16
  float* rs_s = mu_s + 16;        // 16
  float* vn_s = rs_s + 16;        // 16

  for (int i = threadIdx.x; i < 256; i += 64) { Whu_s[i] = Whu[i]; Wuu_s[i] = Wuu[i]; }

  const int lane = threadIdx.x & 31;
  const int lid = lane & 15, hf = lane >> 4;
  int tile = blockIdx.x * 2 + wv;
  const bool wvalid = tile < nTiles;
  if (!wvalid) tile = nTiles - 1;
  const int n0 = tile * 16;

  // h = h_a + s_sum/NORM ; v = v_a + v_sum/NORM
  for (int m = 0; m < 16; ++m) {
    long off = (long)(n0 + m) * 128 + lane * 4;
    const float4 ha = *(const float4*)(h_a + off);
    const float4 ss = *(const float4*)(s_sum + off);
    float* p = sA + m * 148 + lane * 4;
    p[0] = ha.x + ss.x * 0.1f; p[1] = ha.y + ss.y * 0.1f;
    p[2] = ha.z + ss.z * 0.1f; p[3] = ha.w + ss.w * 0.1f;
  }
  {
    long off = (long)(n0 + lid) * 48;
    for (int i = 0; i < 24; ++i) {
      int t = hf * 24 + i;
      vc[lid * 48 + t] = v_a[off + t] + v_sum[off + t] * 0.1f;
    }
  }
  __syncthreads();

  // LN1 stats (half-wave partials)
  {
    float s = 0, s2 = 0;
    for (int i = 0; i < 64; ++i) {
      float x = sA[lid * 148 + hf * 64 + i];
      s += x; s2 += x * x;
    }
    red[lid*4 + hf*2 + 0] = s;
    red[lid*4 + hf*2 + 1] = s2;
  }
  __syncthreads();
  if (hf == 0) {
    float s  = red[lid*4 + 0] + red[lid*4 + 2];
    float s2 = red[lid*4 + 1] + red[lid*4 + 3];
    float mu = s * (1.0f / 128.0f);
    float var = s2 * (1.0f / 128.0f) - mu * mu;
    mu_s[lid] = mu;
    rs_s[lid] = rsqrtf(var + 1e-5f);
  }
  __syncthreads();
  for (int m = 0; m < 16; ++m) {
    float mu = mu_s[m], rs = rs_s[m];
    for (int j = lane; j < 128; j += 32) {
      float x = sA[m * 148 + j];
      sA[m * 148 + j] = (x - mu) * rs * g_msg[j] + b_msg[j];
    }
  }
  // vector norm partials
  {
    float s2 = 0;
    for (int i = 0; i < 24; ++i) { float x = vc[lid*48 + hf*24 + i]; s2 += x * x; }
    red[lid * 2 + hf] = s2;
  }
  __syncthreads();
  if (hf == 0) vn_s[lid] = 1.0f / (sqrtf(red[lid*2] + red[lid*2+1]) + 1e-8f);
  __syncthreads();
  {
    float inv = vn_s[lid];
    for (int i = 0; i < 24; ++i) vc[lid*48 + hf*24 + i] *= inv;
  }
  __syncthreads();

  // update GVP vector path: Vh_u (16ch), sh_u, Vu_u
  for (int h = hf * 8; h < hf * 8 + 8; ++h) {
    float a0 = 0, a1 = 0, a2 = 0;
    for (int k = 0; k < 16; ++k) {
      float w = Whu_s[k * 16 + h];
      a0 += vc[lid*48 + k*3 + 0] * w;
      a1 += vc[lid*48 + k*3 + 1] * w;
      a2 += vc[lid*48 + k*3 + 2] * w;
    }
    vh_s[lid*48 + h*3 + 0] = a0; vh_s[lid*48 + h*3 + 1] = a1; vh_s[lid*48 + h*3 + 2] = a2;
    sA[lid*148 + 128 + h] = sqrtf(a0*a0 + a1*a1 + a2*a2);
  }
  __syncthreads();
  for (int u = hf * 8; u < hf * 8 + 8; ++u) {
    float a0 = 0, a1 = 0, a2 = 0;
    for (int h = 0; h < 16; ++h) {
      float w = Wuu_s[h * 16 + u];
      a0 += vh_s[lid*48 + h*3 + 0] * w;
      a1 += vh_s[lid*48 + h*3 + 1] * w;
      a2 += vh_s[lid*48 + h*3 + 2] * w;
    }
    vu_s[lid*48 + u*3 + 0] = a0; vu_s[lid*48 + u*3 + 1] = a1; vu_s[lid*48 + u*3 + 2] = a2;
  }
  __syncthreads();

  // feats_u GEMM: [16 x 144] @ [144 x 128] (K exact, 36 steps)
  v8f acc[8];
  for (int nt = 0; nt < 8; ++nt) {
    float b0 = bf_u[nt * 16 + lid];
    for (int i = 0; i < 8; ++i) acc[nt][i] = b0;
  }
  for (int ks = 0; ks < 36; ++ks) {
    v2f a;
    int kb = ks * 4 + 2 * hf;
    a[0] = sA[lid * 148 + kb]; a[1] = sA[lid * 148 + kb + 1];
    const float* bp = BpU + (long)ks * 512 + lane * 2;
    for (int nt = 0; nt < 8; ++nt) {
      v2f b; b[0] = bp[nt * 64]; b[1] = bp[nt * 64 + 1];
      acc[nt] = __builtin_amdgcn_wmma_f32_16x16x4_f32(false, a, false, b,
                                                      (short)0, acc[nt], false, false);
    }
  }
  for (int nt = 0; nt < 8; ++nt) {
    for (int r = 0; r < 8; ++r) {
      float x = acc[nt][r];
      int M = r + 8 * hf, N = nt * 16 + lid;
      fo[M * 128 + N] = x / (1.0f + __expf(-x));
    }
  }
  __syncthreads();

  // gate GEMM + v residual
  v8f g;
  { float b0 = bg_u[lid]; for (int i = 0; i < 8; ++i) g[i] = b0; }
  for (int ks = 0; ks < 32; ++ks) {
    v2f a;
    int kb = ks * 4 + 2 * hf;
    a[0] = fo[lid * 128 + kb]; a[1] = fo[lid * 128 + kb + 1];
    const float* bp = BpGU + ks * 64 + lane * 2;
    v2f b; b[0] = bp[0]; b[1] = bp[1];
    g = __builtin_amdgcn_wmma_f32_16x16x4_f32(false, a, false, b, (short)0, g, false, false);
  }
  for (int r = 0; r < 8; ++r) {
    int M = r + 8 * hf;
    float gv = 1.0f / (1.0f + __expf(-g[r]));
    vc[M*48 + lid*3 + 0] += gv * vu_s[M*48 + lid*3 + 0];
    vc[M*48 + lid*3 + 1] += gv * vu_s[M*48 + lid*3 + 1];
    vc[M*48 + lid*3 + 2] += gv * vu_s[M*48 + lid*3 + 2];
  }
  __syncthreads();

  // final LN on h2 = nf + feats_u
  {
    float s = 0, s2 = 0;
    for (int i = 0; i < 64; ++i) {
      float x = sA[lid*148 + hf*64 + i] + fo[lid*128 + hf*64 + i];
      s += x; s2 += x * x;
    }
    red[lid*4 + hf*2 + 0] = s;
    red[lid*4 + hf*2 + 1] = s2;
  }
  __syncthreads();
  if (hf == 0) {
    float s  = red[lid*4 + 0] + red[lid*4 + 2];
    float s2 = red[lid*4 + 1] + red[lid*4 + 3];
    float mu = s * (1.0f / 128.0f);
    float var = s2 * (1.0f / 128.0f) - mu * mu;
    mu_s[lid] = mu;
    rs_s[lid] = rsqrtf(var + 1e-5f);
  }
  __syncthreads();
  // final v norm
  {
    float s2 = 0;
    for (int i = 0; i < 24; ++i) { float x = vc[lid*48 + hf*24 + i]; s2 += x * x; }
    red[lid * 2 + hf] = s2;
  }
  __syncthreads();
  if (hf == 0) vn_s[lid] = 1.0f / (sqrtf(red[lid*2] + red[lid*2+1]) + 1e-8f);
  __syncthreads();

  if (wvalid) {
    for (int m = 0; m < 16; ++m) {
      float mu = mu_s[m], rs = rs_s[m];
      for (int j = lane; j < 128; j += 32) {
        float x = sA[m * 148 + j] + fo[m * 128 + j];
        out[(long)(n0 + m) * 128 + j] = (x - mu) * rs * g_upd[j] + b_upd[j];
      }
    }
    float inv = vn_s[lid];
    long vb = (long)NA * 131 + (long)(n0 + lid) * 48;
    for (int i = 0; i < 24; ++i) {
      int t = hf * 24 + i;
      out[vb + t] = vc[lid * 48 + t] * inv;
    }
    if (hf == 0) {
      long xb = (long)NA * 128 + (long)(n0 + lid) * 3;
      out[xb + 0] = x_a[(long)(n0 + lid) * 3 + 0];
      out[xb + 1] = x_a[(long)(n0 + lid) * 3 + 1];
      out[xb + 2] = x_a[(long)(n0 + lid) * 3 + 2];
    }
  }
}

// ---------------------------------------------------------------------------
extern "C" void kernel_launch(void* const* d_in, const int* in_sizes, int n_in,
                              void* d_out, int out_size, void* d_ws, size_t ws_size,
                              hipStream_t stream) {
  const float* h_a   = (const float*)d_in[0];
  const float* x_a   = (const float*)d_in[1];
  const float* v_a   = (const float*)d_in[2];
  const float* h_b   = (const float*)d_in[3];
  const float* x_b   = (const float*)d_in[4];
  const float* v_b   = (const float*)d_in[5];
  const float* Wh_aa = (const float*)d_in[6];
  const float* Wu_aa = (const float*)d_in[7];
  const float* Wf_aa = (const float*)d_in[8];
  const float* bf_aa = (const float*)d_in[9];
  const float* Wg_aa = (const float*)d_in[10];
  const float* bg_aa = (const float*)d_in[11];
  const float* Wh_ba = (const float*)d_in[12];
  const float* Wu_ba = (const float*)d_in[13];
  const float* Wf_ba = (const float*)d_in[14];
  const float* bf_ba = (const float*)d_in[15];
  const float* Wg_ba = (const float*)d_in[16];
  const float* bg_ba = (const float*)d_in[17];
  const float* Wh_u  = (const float*)d_in[18];
  const float* Wu_u  = (const float*)d_in[19];
  const float* Wf_u  = (const float*)d_in[20];
  const float* bf_u  = (const float*)d_in[21];
  const float* Wg_u  = (const float*)d_in[22];
  const float* bg_u  = (const float*)d_in[23];
  const float* g_msg = (const float*)d_in[24];
  const float* b_msg = (const float*)d_in[25];
  const float* g_upd = (const float*)d_in[26];
  const float* b_upd = (const float*)d_in[27];
  const int* src_aa  = (const int*)d_in[28];
  const int* dst_aa  = (const int*)d_in[29];
  const int* src_ba  = (const int*)d_in[30];
  const int* dst_ba  = (const int*)d_in[31];

  float* ws = (float*)d_ws;
  float* s_sum  = ws;                           // NA*128
  float* v_sum  = s_sum + (size_t)NA * 128;     // NA*48
  float* BpF_aa = v_sum + (size_t)NA * 48;      // 37*512
  float* BpG_aa = BpF_aa + 37 * 512;            // 32*64
  float* BpF_ba = BpG_aa + 32 * 64;
  float* BpG_ba = BpF_ba + 37 * 512;
  float* BpU    = BpG_ba + 32 * 64;             // 36*512
  float* BpGU   = BpU + 36 * 512;               // 32*64

  hipMemsetAsync(s_sum, 0, sizeof(float) * (size_t)NA * 176, stream);

  pack_b_kernel<<<74, 256, 0, stream>>>(Wf_aa, BpF_aa, 145, 128, 8, 37);
  pack_b_kernel<<<8,  256, 0, stream>>>(Wg_aa, BpG_aa, 128, 16, 1, 32);
  pack_b_kernel<<<74, 256, 0, stream>>>(Wf_ba, BpF_ba, 145, 128, 8, 37);
  pack_b_kernel<<<8,  256, 0, stream>>>(Wg_ba, BpG_ba, 128, 16, 1, 32);
  pack_b_kernel<<<72, 256, 0, stream>>>(Wf_u,  BpU,    144, 128, 8, 36);
  pack_b_kernel<<<8,  256, 0, stream>>>(Wg_u,  BpGU,   128, 16, 1, 32);

  const int EDGE_TILES = EE / 16;               // 15625
  const int edge_blocks = (EDGE_TILES + 1) / 2; // 7813
  const size_t edge_lds = (289 + 272 + 2 * 4800) * sizeof(float); // ~39.5 KB
  edge_kernel<<<edge_blocks, 64, edge_lds, stream>>>(
      h_a, v_a, x_a, x_a, src_aa, dst_aa,
      Wh_aa, Wu_aa, BpF_aa, bf_aa, BpG_aa, bg_aa, s_sum, v_sum, EDGE_TILES);
  edge_kernel<<<edge_blocks, 64, edge_lds, stream>>>(
      h_b, v_b, x_b, x_a, src_ba, dst_ba,
      Wh_ba, Wu_ba, BpF_ba, bf_ba, BpG_ba, bg_ba, s_sum, v_sum, EDGE_TILES);

  const int NODE_TILES = NA / 16;               // 6250
  const int node_blocks = NODE_TILES / 2;       // 3125 (exact)
  const size_t node_lds = (512 + 2 * 6832) * sizeof(float); // ~55.7 KB
  node_kernel<<<node_blocks, 64, node_lds, stream>>>(
      h_a, x_a, v_a, s_sum, v_sum,
      Wh_u, Wu_u, BpU, bf_u, BpGU, bg_u,
      g_msg, b_msg, g_upd, b_upd, (float*)d_out, NODE_TILES);
}